// MyGNN_80960133529604
// MI455X (gfx1250) — compile-verified
//
#include <hip/hip_runtime.h>
#include <hip/hip_bf16.h>

typedef float v2f __attribute__((ext_vector_type(2)));
typedef float v8f __attribute__((ext_vector_type(8)));

#define Nn 50000
#define Ee 800000
#define Gg 512
#define Hh 64
#define Ll 5
#define BN_EPS 1e-5f

// ---------------- utility ----------------
__global__ void fill0(float* __restrict__ p, size_t n) {
    size_t i = (size_t)blockIdx.x * blockDim.x + threadIdx.x;
    size_t stride = (size_t)gridDim.x * blockDim.x;
    for (; i < n; i += stride) p[i] = 0.0f;
}

// ---------------- layer 0: scalar scatter, z0 = x + agg0, stats S,Q ----------------
__global__ void scatter_x(const float* __restrict__ x, const int* __restrict__ src,
                          const int* __restrict__ dst, float* __restrict__ agg0, int E) {
    int e = blockIdx.x * blockDim.x + threadIdx.x;
    if (e < E) atomicAdd(&agg0[dst[e]], x[src[e]]);
}

__global__ void layer0_prep(const float* __restrict__ x, const float* __restrict__ agg0,
                            float* __restrict__ z0, float* __restrict__ stats0, int N) {
    int n = blockIdx.x * blockDim.x + threadIdx.x;
    float z = 0.0f, q = 0.0f;
    if (n < N) {
        z = x[n] + agg0[n];
        z0[n] = z;
        q = z * z;
    }
    #pragma unroll
    for (int off = 16; off > 0; off >>= 1) {
        z += __shfl_down(z, off, 32);
        q += __shfl_down(q, off, 32);
    }
    if ((threadIdx.x & 31) == 0) {
        atomicAdd(&stats0[0], z);
        atomicAdd(&stats0[1], q);
    }
}

// scale/shift so that relu(z0*scale[j]+shift[j]) == relu(BN(z0*w1_j + b1_j))
__global__ void finalize0(const float* __restrict__ stats0, const float* __restrict__ w1_0,
                          const float* __restrict__ gm, const float* __restrict__ bm,
                          float* __restrict__ scale, float* __restrict__ shift, float invN) {
    int j = threadIdx.x;  // 64 threads
    float S = stats0[0], Q = stats0[1];
    float meanZ = S * invN;
    float varZ = Q * invN - meanZ * meanZ;
    float w = w1_0[j];
    float inv = rsqrtf(varZ * w * w + BN_EPS);
    float sc = w * inv * gm[j];
    scale[j] = sc;
    shift[j] = bm[j] - meanZ * sc;   // b1 cancels inside (t - mean)
}

// generic BN finalize: mean/var from sums -> affine
__global__ void finalize_bn(const float* __restrict__ stats, const float* __restrict__ g,
                            const float* __restrict__ b, float* __restrict__ scale,
                            float* __restrict__ shift, float invN) {
    int j = threadIdx.x;  // 64 threads
    float mean = stats[j] * invN;
    float var = stats[64 + j] * invN - mean * mean;
    float inv = rsqrtf(var + BN_EPS);
    float sc = g[j] * inv;
    scale[j] = sc;
    shift[j] = b[j] - mean * sc;
}

__global__ void bn_relu_apply(const float* __restrict__ t, const float* __restrict__ scale,
                              const float* __restrict__ shift, float* __restrict__ h, int total) {
    int i = blockIdx.x * blockDim.x + threadIdx.x;
    if (i >= total) return;
    int j = i & 63;
    h[i] = fmaxf(fmaf(t[i], scale[j], shift[j]), 0.0f);
}

// ---------------- vector scatter: one wave per edge, 64 channels ----------------
__global__ void scatter_h(const float* __restrict__ h, const int* __restrict__ src,
                          const int* __restrict__ dst, float* __restrict__ agg, int E) {
    int wid = (blockIdx.x * blockDim.x + threadIdx.x) >> 5;
    int lane = threadIdx.x & 31;
    if (wid >= E) return;
    int s = src[wid], d = dst[wid];
    const float2* hr = (const float2*)(h + (size_t)s * 64);
    float2 v = hr[lane];
    float* ar = agg + (size_t)d * 64 + lane * 2;
    atomicAdd(ar, v.x);
    atomicAdd(ar + 1, v.y);
}

// ---------------- WMMA GEMM: out[N,64] = f(in)[N,64] @ W[64,64] + bias, + BN stats ----------------
// MODE 0: a = in0[n*64+k]                (then affine+optional relu)
// MODE 1: a = in0[n]  (broadcast over k) (then affine+optional relu)
// MODE 2: a = in0[n*64+k] + in1[n*64+k]  (affine is identity via null scale)
template<int MODE>
__global__ void gemm64(const float* __restrict__ in0, const float* __restrict__ in1,
                       const float* __restrict__ colScale, const float* __restrict__ colShift,
                       int applyRelu,
                       const float* __restrict__ W, const float* __restrict__ bias,
                       float* __restrict__ out, float* __restrict__ stats,
                       int N, int tilesTotal) {
    __shared__ float sW[64 * 64];
    __shared__ float sB[64];
    __shared__ float sScale[64];
    __shared__ float sShift[64];
    int tid = threadIdx.x;
    for (int i = tid; i < 64 * 64; i += 256) sW[i] = W[i];
    if (tid < 64) {
        sB[tid] = bias[tid];
        sScale[tid] = colScale ? colScale[tid] : 1.0f;
        sShift[tid] = colShift ? colShift[tid] : 0.0f;
    }
    __syncthreads();

    int wave = tid >> 5;
    int lane = tid & 31;
    int tileId = blockIdx.x * 8 + wave;
    if (tileId >= tilesTotal) return;      // wave-uniform; EXEC stays all-1
    int tm = tileId >> 2;                   // row tile (N/16)
    int tn = tileId & 3;                    // col tile (64/16)
    int hi = lane >> 4;
    int lo = lane & 15;
    int row = tm * 16 + lo;                 // A-matrix row for this lane
    int col = tn * 16 + lo;                 // B/C column for this lane

    float zb = 0.0f;
    if (MODE == 1) zb = in0[row];

    v8f c = {};
    #pragma unroll
    for (int k0 = 0; k0 < 64; k0 += 4) {
        int ka = k0 + 2 * hi;               // this lane's two K slots
        float a0, a1;
        if (MODE == 1) {
            a0 = zb; a1 = zb;
        } else if (MODE == 2) {
            size_t base = (size_t)row * 64 + ka;
            a0 = in0[base] + in1[base];
            a1 = in0[base + 1] + in1[base + 1];
        } else {
            size_t base = (size_t)row * 64 + ka;
            a0 = in0[base];
            a1 = in0[base + 1];
        }
        a0 = fmaf(a0, sScale[ka], sShift[ka]);
        a1 = fmaf(a1, sScale[ka + 1], sShift[ka + 1]);
        if (applyRelu) { a0 = fmaxf(a0, 0.0f); a1 = fmaxf(a1, 0.0f); }
        v2f a; a.x = a0; a.y = a1;
        v2f b; b.x = sW[ka * 64 + col]; b.y = sW[(ka + 1) * 64 + col];
        c = __builtin_amdgcn_wmma_f32_16x16x4_f32(
                /*neg_a=*/false, a, /*neg_b=*/false, b,
                /*c_mod=*/(short)0, c, /*reuse_a=*/false, /*reuse_b=*/false);
    }

    // epilogue: bias add, store, per-column BN statistics
    float bj = sB[col];
    float s = 0.0f, q = 0.0f;
    #pragma unroll
    for (int r = 0; r < 8; ++r) {
        int grow = tm * 16 + r + 8 * hi;    // C/D layout: row = r + 8*(lane>=16)
        float v = c[r] + bj;
        out[(size_t)grow * 64 + col] = v;
        s += v;
        q += v * v;
    }
    s += __shfl_xor(s, 16, 32);             // fold the two 8-row halves of column
    q += __shfl_xor(q, 16, 32);
    if (hi == 0) {
        atomicAdd(&stats[col], s);
        atomicAdd(&stats[64 + col], q);
    }
}

// ---------------- pooling: one block per graph (batch is sorted) ----------------
__global__ void pool_kernel(const float* __restrict__ x, const float* __restrict__ hbase,
                            const int* __restrict__ batch, float* __restrict__ gdesc, int N) {
    __shared__ int sRange[2];
    int g = blockIdx.x;
    if (threadIdx.x == 0) {
        int lo = 0, hb = N;
        while (lo < hb) { int mid = (lo + hb) >> 1; if (batch[mid] < g) lo = mid + 1; else hb = mid; }
        sRange[0] = lo;
        hb = N;
        while (lo < hb) { int mid = (lo + hb) >> 1; if (batch[mid] < g + 1) lo = mid + 1; else hb = mid; }
        sRange[1] = lo;
    }
    __syncthreads();
    int c = threadIdx.x;
    if (c >= 321) return;
    int start = sRange[0], end = sRange[1];
    float acc = 0.0f;
    if (c == 0) {
        for (int n = start; n < end; ++n) acc += x[n];
    } else {
        int l = (c - 1) >> 6, cc = (c - 1) & 63;
        const float* hp = hbase + (size_t)l * Nn * 64 + cc;
        for (int n = start; n < end; ++n) acc += hp[(size_t)n * 64];
    }
    gdesc[g * 321 + c] = acc;
}

__global__ void final_linear(const float* __restrict__ gdesc, const float* __restrict__ lin_w,
                             const float* __restrict__ lin_b, float* __restrict__ out, int G) {
    int t = blockIdx.x * blockDim.x + threadIdx.x;
    if (t >= G * 2) return;
    int g = t >> 1, k = t & 1;
    const float* gd = gdesc + (size_t)g * 321;
    float acc = lin_b[k];
    for (int c = 0; c < 321; ++c) acc = fmaf(gd[c], lin_w[c * 2 + k], acc);
    out[t] = acc;
}

extern "C" void kernel_launch(void* const* d_in, const int* in_sizes, int n_in,
                              void* d_out, int out_size, void* d_ws, size_t ws_size,
                              hipStream_t stream) {
    const float* x       = (const float*)d_in[0];
    const int*   ei      = (const int*)d_in[1];
    const int*   batch   = (const int*)d_in[2];
    const float* w1_0    = (const float*)d_in[3];
    const float* w1_rest = (const float*)d_in[4];
    const float* b1      = (const float*)d_in[5];
    const float* gm      = (const float*)d_in[6];
    const float* bm      = (const float*)d_in[7];
    const float* w2      = (const float*)d_in[8];
    const float* b2      = (const float*)d_in[9];
    const float* go      = (const float*)d_in[10];
    const float* bo      = (const float*)d_in[11];
    const float* lin_w   = (const float*)d_in[12];
    const float* lin_b   = (const float*)d_in[13];
    float* out = (float*)d_out;

    const int N = Nn, E = Ee, G = Gg, H = Hh, L = Ll;
    const int NH = N * H;
    const int* src = ei;
    const int* dst = ei + E;

    // workspace carve-out (≈103 MB)
    char* p = (char*)d_ws;
    auto carve = [&](size_t bytes) { void* r = (void*)p; p += (bytes + 255) & ~(size_t)255; return r; };
    float* hbase = (float*)carve((size_t)L * NH * sizeof(float));   // h_0..h_4
    float* t1    = (float*)carve((size_t)NH * sizeof(float));
    float* t2    = (float*)carve((size_t)NH * sizeof(float));
    float* agg   = (float*)carve((size_t)NH * sizeof(float));       // also agg0 (first N)
    float* z0    = (float*)carve((size_t)N * sizeof(float));
    float* gdesc = (float*)carve((size_t)G * 321 * sizeof(float));
    float* stats = (float*)carve((size_t)(10 * 128 + 2) * sizeof(float));
    float* s1sc  = (float*)carve(64 * sizeof(float));
    float* s1sh  = (float*)carve(64 * sizeof(float));
    float* s2sc  = (float*)carve(64 * sizeof(float));
    float* s2sh  = (float*)carve(64 * sizeof(float));

    const float invN = 1.0f / (float)N;
    const int tiles = (N / 16) * 4;              // 12500 (N % 16 == 0)
    const int gemmBlocks = (tiles + 7) / 8;
    float* stats0 = stats + 10 * 128;

    fill0<<<512, 256, 0, stream>>>(stats, (size_t)(10 * 128 + 2));

    // ---- layer 0 ----
    fill0<<<512, 256, 0, stream>>>(agg, (size_t)N);
    scatter_x<<<(E + 255) / 256, 256, 0, stream>>>(x, src, dst, agg, E);
    layer0_prep<<<(N + 255) / 256, 256, 0, stream>>>(x, agg, z0, stats0, N);
    finalize0<<<1, 64, 0, stream>>>(stats0, w1_0, gm, bm, s1sc, s1sh, invN);
    gemm64<1><<<gemmBlocks, 256, 0, stream>>>(z0, nullptr, s1sc, s1sh, 1,
                                              w2, b2, t2, stats + 5 * 128, N, tiles);
    finalize_bn<<<1, 64, 0, stream>>>(stats + 5 * 128, go, bo, s2sc, s2sh, invN);
    bn_relu_apply<<<(NH + 255) / 256, 256, 0, stream>>>(t2, s2sc, s2sh, hbase, NH);

    // ---- layers 1..4 ----
    for (int l = 1; l < L; ++l) {
        const float* hprev = hbase + (size_t)(l - 1) * NH;
        float* hcur = hbase + (size_t)l * NH;
        float* stA = stats + (size_t)l * 128;
        float* stB = stats + (size_t)(5 + l) * 128;

        fill0<<<2048, 256, 0, stream>>>(agg, (size_t)NH);
        scatter_h<<<(E * 32 + 255) / 256, 256, 0, stream>>>(hprev, src, dst, agg, E);
        gemm64<2><<<gemmBlocks, 256, 0, stream>>>(hprev, agg, nullptr, nullptr, 0,
                                                  w1_rest + (size_t)(l - 1) * 4096,
                                                  b1 + l * 64, t1, stA, N, tiles);
        finalize_bn<<<1, 64, 0, stream>>>(stA, gm + l * 64, bm + l * 64, s1sc, s1sh, invN);
        gemm64<0><<<gemmBlocks, 256, 0, stream>>>(t1, nullptr, s1sc, s1sh, 1,
                                                  w2 + (size_t)l * 4096,
                                                  b2 + l * 64, t2, stB, N, tiles);
        finalize_bn<<<1, 64, 0, stream>>>(stB, go + l * 64, bo + l * 64, s2sc, s2sh, invN);
        bn_relu_apply<<<(NH + 255) / 256, 256, 0, stream>>>(t2, s2sc, s2sh, hcur, NH);
    }

    // ---- readout ----
    pool_kernel<<<G, 352, 0, stream>>>(x, hbase, batch, gdesc, N);
    final_linear<<<(G * 2 + 255) / 256, 256, 0, stream>>>(gdesc, lin_w, lin_b, out, G);
}